// TSFuzzyLayer_4389456576798
// MI455X (gfx1250) — compile-verified
//
#include <hip/hip_runtime.h>

// ---------------------------------------------------------------------------
// TS fuzzy layer for MI455X (gfx1250, wave32).
//   K1: per-edge fuzzy attention; consequence matmul done with
//       V_WMMA_F32_16X16X4_F32 (M=rule, K=premise-dim, N=edge).
//   K2..K6: deterministic 800K softmax (partial max -> max -> partial
//       sum(exp) -> sum -> normalize), tree reductions only (no atomics).
// ---------------------------------------------------------------------------

typedef float v2f __attribute__((ext_vector_type(2)));
typedef float v8f __attribute__((ext_vector_type(8)));

#define N_EDGES_C 3200000
#define N_SG_C    800000
#define NPART     512
#define BLK       256

__global__ __launch_bounds__(BLK) void fuzzy_main_kernel(
    const float4* __restrict__ feat,
    const int*    __restrict__ src,
    const int*    __restrict__ dst,
    const float*  __restrict__ mat,   // 2 x 9 row-major
    const float*  __restrict__ bias,  // 9
    float*        __restrict__ out)
{
    const int e = blockIdx.x * BLK + threadIdx.x;   // E % 256 == 0: no tail
    const int l = threadIdx.x & 31;                 // lane in wave32
    const bool lo = (l < 16);

    // ---- gather + antecedent generation -------------------------------
    float4 fs = feat[src[e]];
    float4 fd = feat[dst[e]];
    float dx = fd.x - fs.x, dy = fd.y - fs.y;
    float vx = fd.z - fs.z, vy = fd.w - fs.w;
    float x1 = sqrtf(dx * dx + dy * dy);
    float vn = sqrtf(vx * vx + vy * vy);
    float cosv = (dx * vx + dy * vy) / (x1 * vn + 1e-8f);
    cosv = fminf(fmaxf(cosv, -1.0f + 1e-6f), 1.0f - 1e-6f);
    float x2 = acosf(cosv) * 57.29577951308232f;    // degrees

    // ---- consequence = [x1,x2] @ SUB_MAT + bias  via WMMA --------------
    // A (16x4) = SUB_MAT^T padded: A[m][0]=mat[0][m], A[m][1]=mat[1][m].
    // ISA A layout: V0 = {lanes0-15:K0, lanes16-31:K2}, V1 = {K1, K3}.
    v2f A;
    A.x = (l < 9) ? mat[l]     : 0.0f;   // K=0 col (lanes>=16 -> K=2 = 0)
    A.y = (l < 9) ? mat[9 + l] : 0.0f;   // K=1 col (lanes>=16 -> K=3 = 0)

    // C (16x16) = bias broadcast along N. VGPR r: lanes0-15 M=r, lanes16-31 M=r+8.
    v8f C;
    C[0] = lo ? bias[0] : bias[8];
#pragma unroll
    for (int r = 1; r < 8; ++r) C[r] = lo ? bias[r] : 0.0f;

    // B (4x16): V0 = {lanes0-15:K0 row, lanes16-31:K2=0}, V1 = {K1, K3=0}.
    v2f B1; B1.x = lo ? x1 : 0.0f; B1.y = lo ? x2 : 0.0f;     // edges 0-15
    float x1s = __shfl_xor(x1, 16);
    float x2s = __shfl_xor(x2, 16);
    v2f B2; B2.x = lo ? x1s : 0.0f; B2.y = lo ? x2s : 0.0f;   // edges 16-31

    v8f D1 = __builtin_amdgcn_wmma_f32_16x16x4_f32(false, A, false, B1,
                                                   (short)0, C, false, false);
    v8f D2 = __builtin_amdgcn_wmma_f32_16x16x4_f32(false, A, false, B2,
                                                   (short)0, C, false, false);

    // D[r] lane n = consequence[rule r][edge n]; rule 8 sits in high lanes of V0.
    float cons[9];
#pragma unroll
    for (int r = 0; r < 8; ++r) {
        float hi = __shfl_xor(D2[r], 16);   // edge l (l>=16) from lane l-16
        cons[r] = lo ? D1[r] : hi;
    }
    {
        float r8lo = __shfl_xor(D1[0], 16); // rule 8, edges 0-15
        cons[8] = lo ? r8lo : D2[0];        // rule 8, edges 16-31: own lane
    }

    // ---- fuzzy truth + weighted average --------------------------------
    // gauss(x,m,s) = exp(-(x-m)^2 / (2 s^2));  2*0.75^2 = 1.125, 2*30^2 = 1800
    float m1v[3], m2v[3];
    {
        float a0 = x1,        a1 = x1 - 2.0f,  a2 = x1 - 4.0f;
        m1v[0] = expf(-(a0 * a0) * (1.0f / 1.125f));
        m1v[1] = expf(-(a1 * a1) * (1.0f / 1.125f));
        m1v[2] = expf(-(a2 * a2) * (1.0f / 1.125f));
        float b0 = x2,        b1 = x2 - 90.0f, b2 = x2 - 180.0f;
        m2v[0] = expf(-(b0 * b0) * (1.0f / 1800.0f));
        m2v[1] = expf(-(b1 * b1) * (1.0f / 1800.0f));
        m2v[2] = expf(-(b2 * b2) * (1.0f / 1800.0f));
    }
    float num = 0.0f, den = 0.0f;
#pragma unroll
    for (int a = 0; a < 3; ++a)
#pragma unroll
        for (int b = 0; b < 3; ++b) {
            float t = fminf(m1v[a], m2v[b]);
            num += t * cons[3 * a + b];
            den += t;
        }
    out[e] = num / den;
}

// ---------------- deterministic softmax over out[idx[0..N_SG)] ------------

__device__ __forceinline__ float block_reduce_max(float v) {
    __shared__ float s[BLK];
    s[threadIdx.x] = v; __syncthreads();
    for (int o = BLK / 2; o > 0; o >>= 1) {
        if (threadIdx.x < o) s[threadIdx.x] = fmaxf(s[threadIdx.x], s[threadIdx.x + o]);
        __syncthreads();
    }
    return s[0];
}
__device__ __forceinline__ float block_reduce_sum(float v) {
    __shared__ float s[BLK];
    s[threadIdx.x] = v; __syncthreads();
    for (int o = BLK / 2; o > 0; o >>= 1) {
        if (threadIdx.x < o) s[threadIdx.x] = s[threadIdx.x] + s[threadIdx.x + o];
        __syncthreads();
    }
    return s[0];
}

__global__ __launch_bounds__(BLK) void pmax_kernel(
    const float* __restrict__ out, const int* __restrict__ idx,
    float* __restrict__ part)
{
    float m = -INFINITY;
    for (int i = blockIdx.x * BLK + threadIdx.x; i < N_SG_C; i += gridDim.x * BLK)
        m = fmaxf(m, out[idx[i]]);
    float r = block_reduce_max(m);
    if (threadIdx.x == 0) part[blockIdx.x] = r;
}

__global__ __launch_bounds__(BLK) void fmax_kernel(
    const float* __restrict__ part, float* __restrict__ g)
{
    float m = -INFINITY;
    for (int i = threadIdx.x; i < NPART; i += BLK) m = fmaxf(m, part[i]);
    float r = block_reduce_max(m);
    if (threadIdx.x == 0) g[0] = r;
}

__global__ __launch_bounds__(BLK) void psum_kernel(
    const float* __restrict__ out, const int* __restrict__ idx,
    const float* __restrict__ g, float* __restrict__ part)
{
    float gmax = g[0];
    float acc = 0.0f;
    for (int i = blockIdx.x * BLK + threadIdx.x; i < N_SG_C; i += gridDim.x * BLK)
        acc += expf(out[idx[i]] - gmax);
    float r = block_reduce_sum(acc);
    if (threadIdx.x == 0) part[blockIdx.x] = r;
}

__global__ __launch_bounds__(BLK) void fsum_kernel(
    const float* __restrict__ part, float* __restrict__ g)
{
    float acc = 0.0f;
    for (int i = threadIdx.x; i < NPART; i += BLK) acc += part[i];
    float r = block_reduce_sum(acc);
    if (threadIdx.x == 0) g[1] = r;
}

__global__ __launch_bounds__(BLK) void norm_kernel(
    float* __restrict__ out, const int* __restrict__ idx,
    const float* __restrict__ g)
{
    int i = blockIdx.x * BLK + threadIdx.x;
    if (i < N_SG_C) {
        int j = idx[i];                       // arange -> permutation, no conflicts
        out[j] = expf(out[j] - g[0]) / g[1];
    }
}

// ---------------------------------------------------------------------------

extern "C" void kernel_launch(void* const* d_in, const int* in_sizes, int n_in,
                              void* d_out, int out_size, void* d_ws, size_t ws_size,
                              hipStream_t stream)
{
    const float4* feat = (const float4*)d_in[0];       // (100000, 4) f32
    const int*    src  = (const int*)d_in[1];          // (E,) i32
    const int*    dst  = (const int*)d_in[2];          // (E,) i32
    const int*    idx  = (const int*)d_in[3];          // (1, N_SG) i32
    const float*  mat  = (const float*)d_in[4];        // (2, 9) f32
    const float*  bias = (const float*)d_in[5];        // (9,) f32
    float*        out  = (float*)d_out;                // (E,1,1) f32

    float* g    = (float*)d_ws;        // g[0] = max, g[1] = sum
    float* part = g + 2;               // NPART partials (reused max then sum)

    const int E = in_sizes[1];         // 3,200,000 (divisible by 256)

    fuzzy_main_kernel<<<E / BLK, BLK, 0, stream>>>(feat, src, dst, mat, bias, out);
    pmax_kernel<<<NPART, BLK, 0, stream>>>(out, idx, part);
    fmax_kernel<<<1, BLK, 0, stream>>>(part, g);
    psum_kernel<<<NPART, BLK, 0, stream>>>(out, idx, g, part);
    fsum_kernel<<<1, BLK, 0, stream>>>(part, g);
    norm_kernel<<<(N_SG_C + BLK - 1) / BLK, BLK, 0, stream>>>(out, idx, g);
}